// Temporal_attention_layer_47270410060059
// MI455X (gfx1250) — compile-verified
//
#include <hip/hip_runtime.h>

// Problem dims (match reference)
static constexpr int Bb = 16;    // batch
static constexpr int Nn = 2048;  // nodes
static constexpr int Ff = 64;    // features
static constexpr int Tt = 32;    // frames

typedef __attribute__((ext_vector_type(2))) float v2f;
typedef __attribute__((ext_vector_type(8))) float v8f;

// ---------------------------------------------------------------------------
// Pass 1: stream x (256 MB) exactly once.
//   R[b,n,t]     = sum_f U3[f] * x[b,n,f,t]          -> Rg
//   Apart[blk,i] = per-block partial of sum_n U1[n]*x[b,n,f,t]  (i = t*64+f)
// One wave handles 8 nodes; lane = t. Block = 8 waves = 64 nodes.
// Fully deterministic: fixed-order wave serialization, no atomics.
// ---------------------------------------------------------------------------
__global__ __launch_bounds__(256) void pass1_stream(
    const float* __restrict__ x, const float* __restrict__ U1,
    const float* __restrict__ U3, float* __restrict__ Rg,
    float* __restrict__ Apart)
{
    __shared__ float accA[Ff * Tt];  // [f][t] layout -> conflict-free lane access

    const int tid   = threadIdx.x;
    const int wave  = tid >> 5;
    const int lane  = tid & 31;
    const int b     = blockIdx.x >> 5;   // N/64 = 32 chunks per batch
    const int chunk = blockIdx.x & 31;
    const int nbase = chunk * 64 + wave * 8;

    for (int i = tid; i < Ff * Tt; i += 256) accA[i] = 0.f;
    __syncthreads();

    // U3 broadcast values (uniform loads -> SGPRs/registers)
    float u3v[Ff];
#pragma unroll
    for (int f = 0; f < Ff; ++f) u3v[f] = U3[f];

    float accf[Ff];
#pragma unroll
    for (int f = 0; f < Ff; ++f) accf[f] = 0.f;

    for (int nn = 0; nn < 8; ++nn) {
        const int n = nbase + nn;
        const float u1 = U1[n];
        const float* xp = x + (((size_t)b * Nn + n) * Ff) * Tt + lane;
        // prefetch next node's tile (global_prefetch_b8)
        __builtin_prefetch(xp + (size_t)Ff * Tt, 0, 0);
        float r = 0.f;
#pragma unroll
        for (int f = 0; f < Ff; ++f) {
            // x is read exactly once across the whole problem: non-temporal,
            // keep L2 clean for U2/R reuse in pass 2.
            const float v = __builtin_nontemporal_load(&xp[f * Tt]);
            r       = fmaf(u3v[f], v, r);
            accf[f] = fmaf(u1,     v, accf[f]);
        }
        Rg[((size_t)b * Nn + n) * Tt + lane] = r;
    }

    // Deterministic cross-wave reduction: serialize the 8 waves in fixed order.
    for (int w = 0; w < 8; ++w) {
        if (wave == w) {
#pragma unroll
            for (int f = 0; f < Ff; ++f)
                accA[f * Tt + lane] += accf[f];
        }
        __syncthreads();
    }

    // Write per-block partial in [t][f] order (coalesced over f); no atomics.
    float* part = Apart + (size_t)blockIdx.x * (Ff * Tt);
    for (int i = tid; i < Ff * Tt; i += 256) {
        const int f = i & (Ff - 1);
        const int t = i >> 6;
        part[i] = accA[f * Tt + t];
    }
}

// ---------------------------------------------------------------------------
// Pass 1b: deterministic fixed-order reduction of the 32 per-block partials
// per batch into A[b,t,f]. 32768 outputs, 32 adds each — negligible traffic.
// ---------------------------------------------------------------------------
__global__ __launch_bounds__(256) void pass1b_reduce(
    const float* __restrict__ Apart, float* __restrict__ Ag)
{
    const int idx = blockIdx.x * 256 + threadIdx.x;  // (b, t*64+f)
    const int b   = idx >> 11;                       // / 2048
    const int i   = idx & 2047;
    float s = 0.f;
#pragma unroll 8
    for (int c = 0; c < 32; ++c)
        s += Apart[(size_t)(b * 32 + c) * (Ff * Tt) + i];
    Ag[(size_t)b * (Ff * Tt) + i] = s;
}

// ---------------------------------------------------------------------------
// Pass 2: M[b,f,s] = sum_n U2[f,n] * R[b,n,s]  via V_WMMA_F32_16X16X4_F32.
// One block per batch (8 waves); one wave per 16x16 tile of the 64x32 output.
// K loop: 2048 in steps of 4.
// A (16x4 f32, 2 VGPRs): lanes 0-15 -> K = {k, k+1}, lanes 16-31 -> {k+2, k+3}
// B (4x16 f32, 2 VGPRs): lanes 0-15 -> rows K=k,k+1; lanes 16-31 -> K=k+2,k+3
// D (16x16 f32, 8 VGPRs): lane col = lane%16, row = vgpr + 8*(lane>=16)
// ---------------------------------------------------------------------------
__global__ __launch_bounds__(256) void pass2_wmma(
    const float* __restrict__ U2, const float* __restrict__ Rg,
    float* __restrict__ Mg)
{
    const int b    = blockIdx.x;          // 16 blocks
    const int wave = threadIdx.x >> 5;    // 8 waves
    const int lane = threadIdx.x & 31;
    const int ft   = wave >> 1;           // f tile: 0..3
    const int st   = wave & 1;            // s tile: 0..1
    const int m    = lane & 15;           // A row / B,D column within tile
    const int hi   = lane >> 4;           // half-wave selector

    const int frow = ft * 16 + m;         // U2 row this lane feeds (A matrix)
    const int scol = st * 16 + m;         // R/M column this lane feeds (B/D)

    const float* u2p = U2 + (size_t)frow * Nn;
    const float* rp  = Rg + (size_t)b * Nn * Tt + scol;

    v8f acc = {};
#pragma unroll 4
    for (int k = 0; k < Nn; k += 4) {
        const int ka = k + hi * 2;
        v2f a;
        a.x = u2p[ka];
        a.y = u2p[ka + 1];
        v2f bm;
        bm.x = rp[(size_t)ka * Tt];
        bm.y = rp[(size_t)(ka + 1) * Tt];
        acc = __builtin_amdgcn_wmma_f32_16x16x4_f32(
            /*neg_a=*/false, a, /*neg_b=*/false, bm,
            /*c_mod=*/(short)0, acc, /*reuse_a=*/false, /*reuse_b=*/false);
    }

#pragma unroll
    for (int r = 0; r < 8; ++r) {
        const int f = ft * 16 + r + hi * 8;
        Mg[((size_t)b * Ff + f) * Tt + scol] = acc[r];
    }
}

// ---------------------------------------------------------------------------
// Pass 3 (tiny): per batch
//   P[t,s] = sum_f A[b,t,f] * M[b,f,s]
//   S      = sigmoid(P + b_e)
//   E[u,s] = sum_t V_e[u,t] * S[t,s]
//   out    = softmax over u (axis=1)
// One 1024-thread block per batch; thread = (u, s).
// ---------------------------------------------------------------------------
__global__ __launch_bounds__(1024) void pass3_finalize(
    const float* __restrict__ Ag, const float* __restrict__ Mg,
    const float* __restrict__ be, const float* __restrict__ Ve,
    float* __restrict__ out)
{
    __shared__ float Al[Tt * Ff];
    __shared__ float Ml[Ff * Tt];
    __shared__ float Sl[Tt * Tt];
    __shared__ float El[Tt * Tt];

    const int b   = blockIdx.x;
    const int tid = threadIdx.x;
    const int s   = tid & 31;
    const int u   = tid >> 5;   // doubles as t for the P stage

    for (int i = tid; i < Tt * Ff; i += 1024) Al[i] = Ag[(size_t)b * Tt * Ff + i];
    for (int i = tid; i < Ff * Tt; i += 1024) Ml[i] = Mg[(size_t)b * Ff * Tt + i];
    __syncthreads();

    float p = 0.f;
#pragma unroll
    for (int f = 0; f < Ff; ++f) p = fmaf(Al[u * Ff + f], Ml[f * Tt + s], p);
    p += be[u * Tt + s];
    const float sg = 1.0f / (1.0f + __expf(-p));
    Sl[u * Tt + s] = sg;
    __syncthreads();

    float e = 0.f;
#pragma unroll
    for (int t = 0; t < Tt; ++t) e = fmaf(Ve[u * Tt + t], Sl[t * Tt + s], e);
    El[u * Tt + s] = e;
    __syncthreads();

    float mx = -3.402823466e38f;
#pragma unroll
    for (int t = 0; t < Tt; ++t) mx = fmaxf(mx, El[t * Tt + s]);
    float sum = 0.f;
#pragma unroll
    for (int t = 0; t < Tt; ++t) sum += __expf(El[t * Tt + s] - mx);

    out[(size_t)b * Tt * Tt + u * Tt + s] = __expf(e - mx) / sum;
}

// ---------------------------------------------------------------------------
extern "C" void kernel_launch(void* const* d_in, const int* in_sizes, int n_in,
                              void* d_out, int out_size, void* d_ws, size_t ws_size,
                              hipStream_t stream) {
    const float* x  = (const float*)d_in[0];
    const float* U1 = (const float*)d_in[1];
    const float* U2 = (const float*)d_in[2];
    const float* U3 = (const float*)d_in[3];
    const float* be = (const float*)d_in[4];
    const float* Ve = (const float*)d_in[5];
    float* out = (float*)d_out;

    // Workspace layout (floats):
    //   Rg    : B*N*T            = 1,048,576
    //   Apart : 512 blocks * T*F = 1,048,576
    //   Ag    : B*T*F            =    32,768
    //   Mg    : B*F*T            =    32,768
    float* Rg    = (float*)d_ws;
    float* Apart = Rg + (size_t)Bb * Nn * Tt;
    float* Ag    = Apart + (size_t)(Bb * 32) * (Ff * Tt);
    float* Mg    = Ag + (size_t)Bb * Tt * Ff;

    pass1_stream<<<Bb * (Nn / 64), 256, 0, stream>>>(x, U1, U3, Rg, Apart);
    pass1b_reduce<<<(Bb * Tt * Ff) / 256, 256, 0, stream>>>(Apart, Ag);
    pass2_wmma<<<Bb, 256, 0, stream>>>(U2, Rg, Mg);
    pass3_finalize<<<Bb, 1024, 0, stream>>>(Ag, Mg, be, Ve, out);
}